// AttentionalPoolingMechanism_8057358648344
// MI455X (gfx1250) — compile-verified
//
#include <hip/hip_runtime.h>
#include <hip/hip_bf16.h>
#include <math.h>

// Problem sizes (compile-time, match reference)
constexpr int BATCH = 64;
constexpr int T_LEN = 8192;
constexpr int FEAT  = 512;
constexpr int HID   = 256;

constexpr int SPLITS = 32;             // workgroups per batch for pass 2
constexpr int TS     = T_LEN / SPLITS; // 256 timesteps per workgroup
constexpr int REC    = 2 + HID;        // partial record: m, d, con[256]

typedef float v2f __attribute__((ext_vector_type(2)));
typedef float v4f __attribute__((ext_vector_type(4)));
typedef float v8f __attribute__((ext_vector_type(8)));

// ---------------------------------------------------------------------------
// Kernel 1: AP = tanh(A @ W^T + bias) via V_WMMA_F32_16X16X4_F32
// One wave per 16x16 output tile. A:[64,512] row-major, W:[256,512] row-major
// (so W^T[k,n] = W[n,k] -> B-matrix columns come from rows of W).
// ---------------------------------------------------------------------------
__global__ __launch_bounds__(32) void proj_wmma_kernel(
    const float* __restrict__ A, const float* __restrict__ W,
    const float* __restrict__ bias, float* __restrict__ AP) {
  const int n0   = blockIdx.x * 16;   // hid tile
  const int m0   = blockIdx.y * 16;   // batch tile
  const int lane = threadIdx.x;       // 0..31, wave32, EXEC all ones
  const int idx  = lane & 15;
  const int half = lane >> 4;         // 0: K={0,1}, 1: K={2,3}

  const float* arow = A + (size_t)(m0 + idx) * FEAT;  // A-matrix row m
  const float* wrow = W + (size_t)(n0 + idx) * FEAT;  // B-matrix col n (= W row)

  v8f c = {0.f, 0.f, 0.f, 0.f, 0.f, 0.f, 0.f, 0.f};
  for (int k0 = 0; k0 < FEAT; k0 += 4) {
    const int ko = k0 + half * 2;
    v2f a = *(const v2f*)(arow + ko);   // K = ko, ko+1 for row m0+idx
    v2f w = *(const v2f*)(wrow + ko);   // K = ko, ko+1 for col n0+idx
    // 8 args: (neg_a, A, neg_b, B, c_mod, C, reuse_a, reuse_b)
    c = __builtin_amdgcn_wmma_f32_16x16x4_f32(false, a, false, w,
                                              (short)0, c, false, false);
  }

  // D layout: VGPR r, lanes 0-15 -> M=r, lanes 16-31 -> M=r+8; N = idx
  const float bb = bias[n0 + idx];
#pragma unroll
  for (int r = 0; r < 8; ++r) {
    const int m = m0 + r + half * 8;
    AP[(size_t)m * HID + (n0 + idx)] = tanhf(c[r] + bb);
  }
}

// ---------------------------------------------------------------------------
// Kernel 2: single-pass online softmax(-s) weighted pooling over a T-slice.
// Grid: (SPLITS, BATCH) x 256 threads (8 waves). Each wave handles one
// timestep per iteration: lane owns HID slice [lane*8, lane*8+8).
// B is 512 MB streamed once (> 192 MB L2): use non-temporal b128 loads.
// Writes partial record {m, d, con[256]} per (batch, split).
// ---------------------------------------------------------------------------
__global__ __launch_bounds__(256) void attn_partial_kernel(
    const float* __restrict__ B, const float* __restrict__ AP,
    float* __restrict__ part) {
  const int s    = blockIdx.x;
  const int b    = blockIdx.y;
  const int tid  = threadIdx.x;
  const int wave = tid >> 5;
  const int lane = tid & 31;

  __shared__ float acc_lds[8 * HID];
  __shared__ float m_lds[8];
  __shared__ float d_lds[8];

  const float* Bb = B + (size_t)b * T_LEN * HID;

  float ap8[8];
  {
    const float* aprow = AP + (size_t)b * HID + lane * 8;
#pragma unroll
    for (int j = 0; j < 8; ++j) ap8[j] = aprow[j];
  }

  float acc[8];
#pragma unroll
  for (int j = 0; j < 8; ++j) acc[j] = 0.f;
  float m = -INFINITY;
  float d = 0.f;

  const int tbase = s * TS;
  for (int i = 0; i < TS / 8; ++i) {
    const int t = tbase + wave + i * 8;
    const float* bv = Bb + (size_t)t * HID + lane * 8;
    // prefetch next iteration's cachelines (global_prefetch_b8, NT)
    __builtin_prefetch(bv + 8 * HID, 0, 0);

    // single-use stream: non-temporal 128-bit loads (TH=NT)
    const v4f v0 = __builtin_nontemporal_load((const v4f*)(bv));
    const v4f v1 = __builtin_nontemporal_load((const v4f*)(bv + 4));

    float dot = ap8[0] * v0.x + ap8[1] * v0.y + ap8[2] * v0.z + ap8[3] * v0.w +
                ap8[4] * v1.x + ap8[5] * v1.y + ap8[6] * v1.z + ap8[7] * v1.w;
    // wave32 all-reduce sum
#pragma unroll
    for (int off = 16; off >= 1; off >>= 1)
      dot += __shfl_xor(dot, off, 32);

    const float x     = -dot;            // reference weight is exp(-s)
    const float mn    = fmaxf(m, x);
    const float scale = __expf(m - mn);  // first iter: exp(-inf) = 0
    const float p     = __expf(x - mn);
    d = d * scale + p;
    acc[0] = acc[0] * scale + p * v0.x;
    acc[1] = acc[1] * scale + p * v0.y;
    acc[2] = acc[2] * scale + p * v0.z;
    acc[3] = acc[3] * scale + p * v0.w;
    acc[4] = acc[4] * scale + p * v1.x;
    acc[5] = acc[5] * scale + p * v1.y;
    acc[6] = acc[6] * scale + p * v1.z;
    acc[7] = acc[7] * scale + p * v1.w;
    m = mn;
  }

  // publish per-wave state
#pragma unroll
  for (int j = 0; j < 8; ++j) acc_lds[wave * HID + lane * 8 + j] = acc[j];
  if (lane == 0) {
    m_lds[wave] = m;
    d_lds[wave] = d;
  }
  __syncthreads();

  // cross-wave merge: thread tid handles hid index h = tid
  float M = m_lds[0];
#pragma unroll
  for (int w = 1; w < 8; ++w) M = fmaxf(M, m_lds[w]);

  float con = 0.f, D = 0.f;
#pragma unroll
  for (int w = 0; w < 8; ++w) {
    const float e = __expf(m_lds[w] - M);
    con += e * acc_lds[w * HID + tid];
    D   += e * d_lds[w];
  }

  float* rec = part + ((size_t)b * SPLITS + s) * REC;
  if (tid == 0) {
    rec[0] = M;
    rec[1] = D;
  }
  rec[2 + tid] = con;
}

// ---------------------------------------------------------------------------
// Kernel 3: merge SPLITS partials per batch; out[b, 0, h] = con_h / denom.
// ---------------------------------------------------------------------------
__global__ __launch_bounds__(256) void attn_final_kernel(
    const float* __restrict__ part, float* __restrict__ out) {
  const int b = blockIdx.x;
  const int h = threadIdx.x;

  __shared__ float m_s[SPLITS];
  __shared__ float d_s[SPLITS];
  if (h < SPLITS) {
    const float* rec = part + ((size_t)b * SPLITS + h) * REC;
    m_s[h] = rec[0];
    d_s[h] = rec[1];
  }
  __syncthreads();

  float M = m_s[0];
#pragma unroll
  for (int s2 = 1; s2 < SPLITS; ++s2) M = fmaxf(M, m_s[s2]);

  float con = 0.f, D = 0.f;
  for (int s2 = 0; s2 < SPLITS; ++s2) {
    const float e = __expf(m_s[s2] - M);
    con += e * part[((size_t)b * SPLITS + s2) * REC + 2 + h];
    D   += e * d_s[s2];
  }
  out[(size_t)b * HID + h] = con / D;
}

// ---------------------------------------------------------------------------
extern "C" void kernel_launch(void* const* d_in, const int* in_sizes, int n_in,
                              void* d_out, int out_size, void* d_ws,
                              size_t ws_size, hipStream_t stream) {
  const float* A    = (const float*)d_in[0];  // [64, 512]
  const float* B    = (const float*)d_in[1];  // [64, 8192, 256]
  const float* W    = (const float*)d_in[2];  // [256, 512]
  const float* bias = (const float*)d_in[3];  // [256]
  float* out = (float*)d_out;                 // [64, 1, 256]

  float* ws   = (float*)d_ws;
  float* AP   = ws;                           // 64*256 floats
  float* part = ws + BATCH * HID;             // 64*32*258 floats

  proj_wmma_kernel<<<dim3(HID / 16, BATCH / 16), 32, 0, stream>>>(A, W, bias, AP);
  attn_partial_kernel<<<dim3(SPLITS, BATCH), 256, 0, stream>>>(B, AP, part);
  attn_final_kernel<<<BATCH, 256, 0, stream>>>(part, out);
}